// RGCN_8924942041218
// MI455X (gfx1250) — compile-verified
//
#include <hip/hip_runtime.h>

typedef __attribute__((ext_vector_type(2))) float v2f;
typedef __attribute__((ext_vector_type(8))) float v8f;

#define N_NODES 100000
#define N_REL   4
#define N_EDGES 160000
#define IN_F    128
#define HID_F   128
#define OUT_F   64
#define MT      4   // M-tiles per wave (B-fragment reuse factor)

// ---------------- utility kernels ----------------

__global__ void zero_f32(float* __restrict__ p, int n) {
  int i = blockIdx.x * blockDim.x + threadIdx.x;
  if (i < n) p[i] = 0.0f;
}

// Count out/in degrees for every relation (as floats, accumulated in place).
__global__ void degree_count(const int* __restrict__ edges,
                             float* __restrict__ rs_out,
                             float* __restrict__ rs_in) {
  int i = blockIdx.x * blockDim.x + threadIdx.x;
  if (i >= N_REL * N_EDGES) return;
  int r = i / N_EDGES;
  int e = i - r * N_EDGES;
  int s = edges[(r * 2 + 0) * N_EDGES + e];
  int d = edges[(r * 2 + 1) * N_EDGES + e];
  atomicAdd(&rs_out[r * N_NODES + s], 1.0f);
  atomicAdd(&rs_in [r * N_NODES + d], 1.0f);
}

// deg -> rsqrt(max(deg,1)) in place
__global__ void deg_to_rsqrt(float* __restrict__ p, int n) {
  int i = blockIdx.x * blockDim.x + threadIdx.x;
  if (i < n) {
    float v = p[i];
    v = v < 1.0f ? 1.0f : v;
    p[i] = rsqrtf(v);
  }
}

// Wt[r][n][k] = W[r][k][n]  (makes B fragments contiguous in K)
__global__ void transpose_w(const float* __restrict__ W, float* __restrict__ Wt,
                            int K, int N) {
  int i = blockIdx.x * blockDim.x + threadIdx.x;
  if (i >= N_REL * K * N) return;
  int r   = i / (K * N);
  int rem = i - r * (K * N);
  int k   = rem / N;
  int n   = rem - k * N;
  Wt[(size_t)r * K * N + (size_t)n * K + k] = W[i];
}

// ---------------- WMMA GEMM: H = diag(rs) * (X @ W) ----------------
// X: M x K row major, Wt: N x K row major (pre-transposed), H: M x N.
// K is a compile-time constant -> the 32-step k-loop fully unrolls and the
// scheduler can hoist global_load_b64 fragments several k-steps ahead of the
// consuming v_wmma_f32_16x16x4_f32 (partial s_wait_loadcnt instead of 0x0).
// Each wave computes MT stacked 16x16 C tiles sharing one N-tile, so each
// B fragment feeds MT WMMAs. rsqrt(deg_out) row scaling commutes through the
// GEMM -> applied at the C store.
// A 16x4 f32 layout: lanes 0-15 = K {k,k+1} of row M=lane; lanes 16-31 =
// K {k+2,k+3} of row M=lane-16. B 4x16 mirrored. C/D: VGPR v holds row
// tm+v (lanes 0-15) / tm+v+8 (lanes 16-31), col tn+(lane&15).
template <int K>
__global__ void __launch_bounds__(256)
rgcn_gemm_wmma(const float* __restrict__ X, const float* __restrict__ rs,
               const float* __restrict__ Wt, float* __restrict__ H,
               int M, int N) {
  const int wave   = threadIdx.x >> 5;
  const int lane   = threadIdx.x & 31;
  const int ntiles = N >> 4;
  const int mtiles = M >> 4;
  const int groups = (mtiles + MT - 1) / MT;
  const int w      = blockIdx.x * (blockDim.x >> 5) + wave;   // wave-uniform
  if (w >= groups * ntiles) return;  // uniform -> EXEC stays all-ones

  const int tn    = (w % ntiles) << 4;
  const int tmb   = (w / ntiles) * (MT << 4);   // base row of this tile group
  const int hi    = lane >> 4;                  // K-pair select
  const int l15   = lane & 15;
  const int n     = tn + l15;
  const int kb0   = hi << 1;

  // A-row pointers (clamped so tail-group loads stay in bounds; stores guarded)
  const float* __restrict__ xrow[MT];
#pragma unroll
  for (int t = 0; t < MT; ++t) {
    int m = tmb + t * 16 + l15;
    if (m > M - 1) m = M - 1;
    xrow[t] = X + (size_t)m * K;
  }
  const float* __restrict__ wrow = Wt + (size_t)n * K;

  v8f acc[MT] = {};

  // One-step software pipeline over the fully unrolled k-loop.
  v2f b_cur = *(const v2f*)(wrow + kb0);
  v2f a_cur[MT];
#pragma unroll
  for (int t = 0; t < MT; ++t) a_cur[t] = *(const v2f*)(xrow[t] + kb0);

#pragma unroll
  for (int k = 4; k < K; k += 4) {
    const int kn = k + kb0;
    const v2f b_nxt = *(const v2f*)(wrow + kn);
    v2f a_nxt[MT];
#pragma unroll
    for (int t = 0; t < MT; ++t) a_nxt[t] = *(const v2f*)(xrow[t] + kn);
#pragma unroll
    for (int t = 0; t < MT; ++t)
      acc[t] = __builtin_amdgcn_wmma_f32_16x16x4_f32(
          false, a_cur[t], false, b_cur, (short)0, acc[t], false, false);
    b_cur = b_nxt;
#pragma unroll
    for (int t = 0; t < MT; ++t) a_cur[t] = a_nxt[t];
  }
#pragma unroll
  for (int t = 0; t < MT; ++t)
    acc[t] = __builtin_amdgcn_wmma_f32_16x16x4_f32(
        false, a_cur[t], false, b_cur, (short)0, acc[t], false, false);

#pragma unroll
  for (int t = 0; t < MT; ++t) {
    const int tm = tmb + t * 16;
    if (tm >= M) break;                               // wave-uniform guard
    float* __restrict__ out = H + (size_t)(tm + hi * 8) * N + tn + l15;
#pragma unroll
    for (int v = 0; v < 8; ++v)
      out[(size_t)v * N] = acc[t][v] * rs[tm + hi * 8 + v];
  }
}

// ---------------- edge scatter: accum[dst] += h[src] * rs_in[dst] ----------------
__global__ void scatter_add(const float* __restrict__ H,
                            const int* __restrict__ src,
                            const int* __restrict__ dst,
                            const float* __restrict__ rs_in,
                            float* __restrict__ accum, int C4) {
  int i = blockIdx.x * blockDim.x + threadIdx.x;
  if (i >= N_EDGES * C4) return;
  int e = i / C4;
  int c = i - e * C4;
  int s = src[e];
  int d = dst[e];
  float sc = rs_in[d];
  const float4 v = ((const float4*)(H + (size_t)s * (C4 * 4)))[c];
  float* ap = accum + (size_t)d * (C4 * 4) + c * 4;
  atomicAdd(ap + 0, v.x * sc);
  atomicAdd(ap + 1, v.y * sc);
  atomicAdd(ap + 2, v.z * sc);
  atomicAdd(ap + 3, v.w * sc);
}

// ---------------- finalize: out = [relu](accum + sum_r b[r]) ----------------
__global__ void finalize(const float* __restrict__ accum,
                         const float* __restrict__ b,
                         float* __restrict__ out, int C, int relu) {
  int i = blockIdx.x * blockDim.x + threadIdx.x;
  if (i >= N_NODES * C) return;
  int c = i % C;
  float bsum = b[c] + b[C + c] + b[2 * C + c] + b[3 * C + c];
  float v = accum[i] + bsum;
  if (relu) v = v > 0.0f ? v : 0.0f;
  out[i] = v;
}

// ---------------- launch ----------------

extern "C" void kernel_launch(void* const* d_in, const int* in_sizes, int n_in,
                              void* d_out, int out_size, void* d_ws, size_t ws_size,
                              hipStream_t stream) {
  const float* x     = (const float*)d_in[0];
  const int*   edges = (const int*)  d_in[1];
  const float* W1    = (const float*)d_in[2];
  const float* b1    = (const float*)d_in[3];
  const float* W2    = (const float*)d_in[4];
  const float* b2    = (const float*)d_in[5];
  float* out = (float*)d_out;

  float* ws     = (float*)d_ws;
  float* rs_out = ws;                               // N_REL * N_NODES
  float* rs_in  = rs_out + N_REL * N_NODES;         // N_REL * N_NODES
  float* htmp   = rs_in  + N_REL * N_NODES;         // N_NODES * HID_F (reused)
  float* accum  = htmp   + (size_t)N_NODES * HID_F; // N_NODES * HID_F (reused)
  float* hidden = accum  + (size_t)N_NODES * HID_F; // N_NODES * HID_F
  float* wt     = hidden + (size_t)N_NODES * HID_F; // N_REL * 128 * 128 (reused)

  const int B = 256;
  auto g = [](long long n, int b) { return (unsigned)((n + b - 1) / b); };

  // --- degrees (shared by both layers) ---
  zero_f32<<<g(2LL * N_REL * N_NODES, B), B, 0, stream>>>(rs_out, 2 * N_REL * N_NODES);
  degree_count<<<g((long long)N_REL * N_EDGES, B), B, 0, stream>>>(edges, rs_out, rs_in);
  deg_to_rsqrt<<<g(2LL * N_REL * N_NODES, B), B, 0, stream>>>(rs_out, 2 * N_REL * N_NODES);

  // --- layer 1: 128 -> 128, relu ---
  transpose_w<<<g((long long)N_REL * IN_F * HID_F, B), B, 0, stream>>>(W1, wt, IN_F, HID_F);
  zero_f32<<<g((long long)N_NODES * HID_F, B), B, 0, stream>>>(accum, N_NODES * HID_F);
  {
    const int mtiles = N_NODES / 16;
    const int ntiles = HID_F / 16;
    const long long waves = (long long)((mtiles + MT - 1) / MT) * ntiles;
    const int C4 = HID_F / 4;
    for (int r = 0; r < N_REL; ++r) {
      rgcn_gemm_wmma<IN_F><<<g(waves, 8), B, 0, stream>>>(
          x, rs_out + r * N_NODES, wt + (size_t)r * IN_F * HID_F, htmp,
          N_NODES, HID_F);
      scatter_add<<<g((long long)N_EDGES * C4, B), B, 0, stream>>>(
          htmp, edges + (r * 2 + 0) * N_EDGES, edges + (r * 2 + 1) * N_EDGES,
          rs_in + r * N_NODES, accum, C4);
    }
  }
  finalize<<<g((long long)N_NODES * HID_F, B), B, 0, stream>>>(accum, b1, hidden, HID_F, 1);

  // --- layer 2: 128 -> 64, no relu ---
  transpose_w<<<g((long long)N_REL * HID_F * OUT_F, B), B, 0, stream>>>(W2, wt, HID_F, OUT_F);
  zero_f32<<<g((long long)N_NODES * OUT_F, B), B, 0, stream>>>(accum, N_NODES * OUT_F);
  {
    const int mtiles = N_NODES / 16;
    const int ntiles = OUT_F / 16;
    const long long waves = (long long)((mtiles + MT - 1) / MT) * ntiles;
    const int C4 = OUT_F / 4;
    for (int r = 0; r < N_REL; ++r) {
      rgcn_gemm_wmma<HID_F><<<g(waves, 8), B, 0, stream>>>(
          hidden, rs_out + r * N_NODES, wt + (size_t)r * HID_F * OUT_F, htmp,
          N_NODES, OUT_F);
      scatter_add<<<g((long long)N_EDGES * C4, B), B, 0, stream>>>(
          htmp, edges + (r * 2 + 0) * N_EDGES, edges + (r * 2 + 1) * N_EDGES,
          rs_in + r * N_NODES, accum, C4);
    }
  }
  finalize<<<g((long long)N_NODES * OUT_F, B), B, 0, stream>>>(accum, b2, out, OUT_F, 0);
}